// NormalizedDynamicsCorrected_57561151701124
// MI455X (gfx1250) — compile-verified
//
#include <hip/hip_runtime.h>
#include <hip/hip_bf16.h>
#include <math.h>

// ---------------------------------------------------------------------------
// NormalizedDynamicsCorrected for MI455X (gfx1250, wave32, WMMA)
// N = 8192 points, D = 64 dims, k = 15, step = 64^-1
// ---------------------------------------------------------------------------

#define NPTS 8192
#define NDIM 64
#define KNN  15
#define NTILE (NPTS / 16)   // 512 column tiles of 16

typedef __attribute__((ext_vector_type(2))) float v2f;
typedef __attribute__((ext_vector_type(8))) float v8f;

// ---------------------------------------------------------------------------
// Kernel 1/4: per-column mean + unbiased std (64 columns -> 64 blocks)
// ---------------------------------------------------------------------------
__global__ __launch_bounds__(256)
void nd_colstats(const float* __restrict__ in,
                 float* __restrict__ meanOut,
                 float* __restrict__ stdOut) {
  __shared__ float s1[256];
  __shared__ float s2[256];
  const int c = blockIdx.x;
  const int t = threadIdx.x;
  float sum = 0.0f, ss = 0.0f;
  for (int i = t; i < NPTS; i += 256) {
    float v = in[(size_t)i * NDIM + c];
    sum += v;
    ss  += v * v;
  }
  s1[t] = sum; s2[t] = ss;
  __syncthreads();
  for (int off = 128; off > 0; off >>= 1) {
    if (t < off) { s1[t] += s1[t + off]; s2[t] += s2[t + off]; }
    __syncthreads();
  }
  if (t == 0) {
    float mean = s1[0] / (float)NPTS;
    float var  = (s2[0] - (float)NPTS * mean * mean) / (float)(NPTS - 1);
    meanOut[c] = mean;
    stdOut[c]  = sqrtf(fmaxf(var, 0.0f));
  }
}

// ---------------------------------------------------------------------------
// Kernel 2: center x and compute per-row sum of squares.
// 4 rows per 256-thread block (thread = row*64 + col), coalesced.
// ---------------------------------------------------------------------------
__global__ __launch_bounds__(256)
void nd_center(const float* __restrict__ x,
               const float* __restrict__ mean,
               float* __restrict__ xc,
               float* __restrict__ sq) {
  __shared__ float red[256];
  const int t   = threadIdx.x;
  const int row = blockIdx.x * 4 + (t >> 6);
  const int c   = t & 63;
  float v = x[(size_t)row * NDIM + c] - mean[c];
  xc[(size_t)row * NDIM + c] = v;
  red[t] = v * v;
  __syncthreads();
  for (int off = 32; off > 0; off >>= 1) {
    if (c < off) red[t] += red[t + off];
    __syncthreads();
  }
  if (c == 0) sq[row] = red[t];
}

// ---------------------------------------------------------------------------
// fp32 WMMA 16x16xK=64 Gram tile: g[m][n] = sum_k A[m][k] * B[n][k]
// A-frag (16x4 f32): lane m holds row m; VGPR0/1 = K {0,1} (lanes 0-15)
//                    or K {2,3} (lanes 16-31) within each k-step of 4.
// B-frag mirrored with lane = N.
// ---------------------------------------------------------------------------
__device__ __forceinline__
v8f nd_gram16(const float A[16][68], const float B[16][68], int lane) {
  const int m   = lane & 15;
  const int hi2 = (lane >> 4) << 1;   // 0 or 2
  v8f c = {};
#pragma unroll
  for (int k0 = 0; k0 < NDIM; k0 += 4) {
    v2f a, b;
    a.x = A[m][k0 + hi2];
    a.y = A[m][k0 + hi2 + 1];
    b.x = B[m][k0 + hi2];
    b.y = B[m][k0 + hi2 + 1];
    c = __builtin_amdgcn_wmma_f32_16x16x4_f32(false, a, false, b,
                                              (short)0, c, false, false);
  }
  return c;
}

// ---------------------------------------------------------------------------
// Kernel 3: main fused kernel. One wave (32 threads) per 16-row tile.
// Pass 1: sigma = 15th-smallest distance per row (top-15 in registers).
// Pass 2: recompute distances, w = exp(-d/(2 sigma^2)); accumulate row sums
//         and drift = w @ xc via WMMA (w staged through LDS C->A relayout).
// Writes h' = xc + step*(drift/rowsum - xc) to hout.
// ---------------------------------------------------------------------------
__global__ __launch_bounds__(32)
void nd_main(const float* __restrict__ xc,
             const float* __restrict__ sq,
             float* __restrict__ hout) {
  __shared__ alignas(16) float A[16][68];   // row tile of xc (padded: 4m+k banks)
  __shared__ alignas(16) float B[16][68];   // column tile of xc
  __shared__ float Wt[16][68];              // w tile, read back as A-fragments
  __shared__ float Dt[16][17];              // distance tile for top-k scan
  __shared__ float part[16][16];            // row-sum partials
  __shared__ float sqA[16], sqB[16], sigInv[16], rowSum[16];

  const int lane = threadIdx.x;
  const int i0   = blockIdx.x * 16;
  const int m    = lane & 15;
  const int hi   = lane >> 4;
  const int hi2  = hi << 1;

  // stage A tile (16 rows x 64 floats) via float4
  for (int t = lane; t < 16 * 16; t += 32) {
    int r = t >> 4, q = t & 15;
    reinterpret_cast<float4*>(&A[r][0])[q] =
        reinterpret_cast<const float4*>(xc + (size_t)(i0 + r) * NDIM)[q];
  }
  if (lane < 16) sqA[lane] = sq[i0 + lane];
  __syncthreads();

  // ---------------- pass 1: per-row top-15 -> sigma ----------------
  float tk[KNN];
#pragma unroll
  for (int a = 0; a < KNN; ++a) tk[a] = 3.0e38f;

  for (int j = 0; j < NTILE; ++j) {
    const int j0 = j * 16;
    __syncthreads();                       // previous readers of B done
    for (int t = lane; t < 16 * 16; t += 32) {
      int r = t >> 4, q = t & 15;
      reinterpret_cast<float4*>(&B[r][0])[q] =
          reinterpret_cast<const float4*>(xc + (size_t)(j0 + r) * NDIM)[q];
    }
    if (lane < 16) sqB[lane] = sq[j0 + lane];
    if (j + 1 < NTILE)
      __builtin_prefetch(xc + (size_t)(j0 + 16) * NDIM + lane * 2, 0, 0);
    __syncthreads();

    v8f g = nd_gram16(A, B, lane);
#pragma unroll
    for (int r = 0; r < 8; ++r) {
      const int row = r + hi * 8;
      float d2 = sqA[row] + sqB[m] - 2.0f * g[r];
      Dt[row][m] = (d2 > 0.0f) ? sqrtf(d2) : 0.0f;
    }
    __syncthreads();
    if (lane < 16) {
#pragma unroll
      for (int n = 0; n < 16; ++n) {
        float d = Dt[lane][n];
        if (d < tk[KNN - 1]) {             // amortized O(1): rare insert
#pragma unroll
          for (int a = 0; a < KNN; ++a) {  // sorted insert, eject largest
            float lo = fminf(tk[a], d);
            d = fmaxf(tk[a], d);
            tk[a] = lo;
          }
        }
      }
    }
  }
  if (lane < 16) {
    float s = tk[KNN - 1];                 // 15th smallest (includes self=0)
    sigInv[lane] = 1.0f / (2.0f * s * s);
  }
  __syncthreads();

  // ---------------- pass 2: kernel weights + drift ----------------
  v8f acc[4];
#pragma unroll
  for (int nt = 0; nt < 4; ++nt) { v8f z = {}; acc[nt] = z; }
  float rs[8];
#pragma unroll
  for (int r = 0; r < 8; ++r) rs[r] = 0.0f;

  for (int j = 0; j < NTILE; ++j) {
    const int j0 = j * 16;
    __syncthreads();                       // drift readers of B/Wt done
    for (int t = lane; t < 16 * 16; t += 32) {
      int r = t >> 4, q = t & 15;
      reinterpret_cast<float4*>(&B[r][0])[q] =
          reinterpret_cast<const float4*>(xc + (size_t)(j0 + r) * NDIM)[q];
    }
    if (lane < 16) sqB[lane] = sq[j0 + lane];
    __syncthreads();

    v8f g = nd_gram16(A, B, lane);
#pragma unroll
    for (int r = 0; r < 8; ++r) {
      const int row = r + hi * 8;
      float d2 = sqA[row] + sqB[m] - 2.0f * g[r];
      float d  = (d2 > 0.0f) ? sqrtf(d2) : 0.0f;
      float w  = __expf(-d * sigInv[row]);
      rs[r] += w;
      Wt[row][m] = w;                      // stage for A-fragment relayout
    }
    __syncthreads();

    // drift tile: acc[nt](16x16) += W(16x16[K]) x B(16[K] x 64)
#pragma unroll
    for (int nt = 0; nt < 4; ++nt) {
#pragma unroll
      for (int k0 = 0; k0 < 16; k0 += 4) {
        v2f a, b;
        a.x = Wt[m][k0 + hi2];
        a.y = Wt[m][k0 + hi2 + 1];
        b.x = B[k0 + hi2][nt * 16 + m];
        b.y = B[k0 + hi2 + 1][nt * 16 + m];
        acc[nt] = __builtin_amdgcn_wmma_f32_16x16x4_f32(false, a, false, b,
                                                        (short)0, acc[nt],
                                                        false, false);
      }
    }
  }

  // reduce per-row kernel sums across the 16 lanes of each half-wave
#pragma unroll
  for (int r = 0; r < 8; ++r) part[r + hi * 8][m] = rs[r];
  __syncthreads();
  if (lane < 16) {
    float s = 0.0f;
    for (int n = 0; n < 16; ++n) s += part[lane][n];
    rowSum[lane] = s;
  }
  __syncthreads();

  const float step = 1.0f / (float)NDIM;   // DIM^-ALPHA
#pragma unroll
  for (int nt = 0; nt < 4; ++nt) {
#pragma unroll
    for (int r = 0; r < 8; ++r) {
      const int row = r + hi * 8;
      const int dim = nt * 16 + m;
      float drift = acc[nt][r] / rowSum[row];
      float xcv   = A[row][dim];
      hout[(size_t)(i0 + row) * NDIM + dim] = xcv + step * (drift - xcv);
    }
  }
}

// ---------------------------------------------------------------------------
// Kernel 5: h = h' * (orig_std / std(h')) + orig_mean  (in place on d_out)
// ---------------------------------------------------------------------------
__global__ __launch_bounds__(256)
void nd_finalize(float* __restrict__ h,
                 const float* __restrict__ std0,
                 const float* __restrict__ hstd,
                 const float* __restrict__ mean0) {
  const int idx = blockIdx.x * 256 + threadIdx.x;
  const int c = idx & 63;
  h[idx] = h[idx] * (std0[c] / hstd[c]) + mean0[c];
}

// ---------------------------------------------------------------------------
// Launch
// ---------------------------------------------------------------------------
extern "C" void kernel_launch(void* const* d_in, const int* in_sizes, int n_in,
                              void* d_out, int out_size, void* d_ws, size_t ws_size,
                              hipStream_t stream) {
  (void)in_sizes; (void)n_in; (void)out_size; (void)ws_size;
  const float* x = (const float*)d_in[0];      // [8192, 64] f32
  float* out = (float*)d_out;                  // [8192, 64] f32

  // workspace layout (floats)
  float* ws    = (float*)d_ws;
  float* xc    = ws;                           // 8192*64
  float* sq    = xc + (size_t)NPTS * NDIM;     // 8192
  float* mean0 = sq + NPTS;                    // 64
  float* std0  = mean0 + NDIM;                 // 64
  float* hmean = std0 + NDIM;                  // 64
  float* hstd  = hmean + NDIM;                 // 64

  nd_colstats<<<NDIM, 256, 0, stream>>>(x, mean0, std0);
  nd_center  <<<NPTS / 4, 256, 0, stream>>>(x, mean0, xc, sq);
  nd_main    <<<NTILE, 32, 0, stream>>>(xc, sq, out);
  nd_colstats<<<NDIM, 256, 0, stream>>>(out, hmean, hstd);
  nd_finalize<<<(NPTS * NDIM) / 256, 256, 0, stream>>>(out, std0, hstd, mean0);
}